// MyModel_61933428413390
// MI455X (gfx1250) — compile-verified
//
#include <hip/hip_runtime.h>
#include <stdint.h>

// out = x + U[0,1) with JAX threefry2x32, key = jax.random.key(42) -> {0, 42}.
// Memory-bound streaming op: ~1.07 GB traffic -> ~46 us floor at 23.3 TB/s.
// Cipher cost ~70 VALU/pair is irreducible; unroll 8 independent pairs per
// iteration for VOPD dual-issue, rotates via v_alignbit_b32, NT B128 traffic.

typedef __attribute__((ext_vector_type(4))) float v4f;

__device__ __forceinline__ uint32_t rotl32(uint32_t v, int r) {
  return __builtin_rotateleft32(v, (uint32_t)r);   // v_alignbit_b32
}

// Threefry-2x32, 20 rounds, exactly JAX's schedule, key {0, 42}.
__device__ __forceinline__ void threefry2x32(uint32_t x0, uint32_t x1,
                                             uint32_t& o0, uint32_t& o1) {
  const uint32_t ks0 = 0u;
  const uint32_t ks1 = 42u;
  const uint32_t ks2 = ks0 ^ ks1 ^ 0x1BD11BDAu;
  x0 += ks0; x1 += ks1;
#define TF_ROUND(r) { x0 += x1; x1 = rotl32(x1, (r)); x1 ^= x0; }
  TF_ROUND(13) TF_ROUND(15) TF_ROUND(26) TF_ROUND(6)
  x0 += ks1; x1 += ks2 + 1u;
  TF_ROUND(17) TF_ROUND(29) TF_ROUND(16) TF_ROUND(24)
  x0 += ks2; x1 += ks0 + 2u;
  TF_ROUND(13) TF_ROUND(15) TF_ROUND(26) TF_ROUND(6)
  x0 += ks0; x1 += ks1 + 3u;
  TF_ROUND(17) TF_ROUND(29) TF_ROUND(16) TF_ROUND(24)
  x0 += ks1; x1 += ks2 + 4u;
  TF_ROUND(13) TF_ROUND(15) TF_ROUND(26) TF_ROUND(6)
  x0 += ks2; x1 += ks0 + 5u;
#undef TF_ROUND
  o0 = x0; o1 = x1;
}

// JAX uniform [0,1): bitcast((bits >> 9) | 0x3f800000) - 1.0f
__device__ __forceinline__ float bits_to_unit_float(uint32_t bits) {
  return __uint_as_float((bits >> 9) | 0x3F800000u) - 1.0f;
}

__global__ __launch_bounds__(256, 1) void add_uniform_tf2x32(
    const float* __restrict__ x, float* __restrict__ out,
    uint32_t groups,   // number of 8-pair groups = (n/2)/8
    uint32_t half) {   // n/2 (elements)
  const uint32_t stride = gridDim.x * blockDim.x;   // in groups
  const uint32_t tid0   = blockIdx.x * blockDim.x + threadIdx.x;

  const v4f* __restrict__ x0v = (const v4f*)x;            // first-half stream
  const v4f* __restrict__ x1v = (const v4f*)(x + half);   // second-half stream
  v4f* __restrict__ o0v = (v4f*)out;
  v4f* __restrict__ o1v = (v4f*)(out + half);

  for (uint32_t g = tid0; g < groups; g += stride) {
    const uint32_t v = g * 2u;               // float4 index within a half
    // Stream-prefetch one grid-stride ahead on both halves (global_prefetch_b8).
    __builtin_prefetch((const void*)(x0v + (v + stride * 2u)), 0, 0);
    __builtin_prefetch((const void*)(x1v + (v + stride * 2u)), 0, 0);

    // Nontemporal B128 loads: read-once data.
    v4f a0 = __builtin_nontemporal_load(x0v + v);
    v4f a1 = __builtin_nontemporal_load(x0v + v + 1u);
    v4f b0 = __builtin_nontemporal_load(x1v + v);
    v4f b1 = __builtin_nontemporal_load(x1v + v + 1u);

    // 8 independent cipher pairs -> 16 independent VALU chains for dual-issue.
    const uint32_t p = g * 8u;               // element index in first half
    uint32_t r0[8], r1[8];
#pragma unroll
    for (int k = 0; k < 8; ++k) {
      // JAX pairing: counters (p, p + n/2); o0 -> out[p], o1 -> out[p + n/2]
      threefry2x32(p + (uint32_t)k, p + (uint32_t)k + half, r0[k], r1[k]);
    }

    v4f c0, c1, c2, c3;
#pragma unroll
    for (int k = 0; k < 4; ++k) {
      c0[k] = a0[k] + bits_to_unit_float(r0[k]);
      c1[k] = a1[k] + bits_to_unit_float(r0[k + 4]);
      c2[k] = b0[k] + bits_to_unit_float(r1[k]);
      c3[k] = b1[k] + bits_to_unit_float(r1[k + 4]);
    }

    // Nontemporal B128 stores: write-once output.
    __builtin_nontemporal_store(c0, o0v + v);
    __builtin_nontemporal_store(c1, o0v + v + 1u);
    __builtin_nontemporal_store(c2, o1v + v);
    __builtin_nontemporal_store(c3, o1v + v + 1u);
  }

  // Scalar tail for half % 8 != 0 (zero-cost for the reference shape).
  const uint32_t rem = half - groups * 8u;
  if (tid0 < rem) {
    const uint32_t p = groups * 8u + tid0;
    uint32_t r0, r1;
    threefry2x32(p, p + half, r0, r1);
    out[p]        = x[p]        + bits_to_unit_float(r0);
    out[p + half] = x[p + half] + bits_to_unit_float(r1);
  }
}

extern "C" void kernel_launch(void* const* d_in, const int* in_sizes, int n_in,
                              void* d_out, int out_size, void* d_ws, size_t ws_size,
                              hipStream_t stream) {
  (void)n_in; (void)out_size; (void)d_ws; (void)ws_size;
  const float* x  = (const float*)d_in[0];
  float* out      = (float*)d_out;

  const uint32_t n      = (uint32_t)in_sizes[0];   // 134,217,728 for the reference
  const uint32_t half   = n / 2u;
  const uint32_t groups = half / 8u;               // 8-pair groups (2 float4s/half)

  const int block = 256;                            // 8 wave32 waves per block
  // ~8 group-iterations per thread -> 4096 blocks for the reference shape
  uint32_t grid = (groups + (uint32_t)block * 8u - 1u) / ((uint32_t)block * 8u);
  if (grid < 1u) grid = 1u;
  if (grid > 65535u) grid = 65535u;

  add_uniform_tf2x32<<<dim3(grid), dim3(block), 0, stream>>>(x, out, groups, half);
}